// _fasterRCNN_54477365182933
// MI455X (gfx1250) — compile-verified
//
#include <hip/hip_runtime.h>
#include <math.h>
#include <stdint.h>
#include <stddef.h>

typedef __attribute__((ext_vector_type(4))) float v4f;

#define THRESH 0.05f

// ---------------------------------------------------------------------------
// Kernel A: decode + clip.  One thread per (box, class) float4.
// deltas/pred are streamed non-temporally; boxes hit L0/L2 (7x reuse).
// ---------------------------------------------------------------------------
__global__ __launch_bounds__(256)
void decode_clip_kernel(const v4f* __restrict__ deltas4,
                        const v4f* __restrict__ boxes4,
                        const float* __restrict__ im_info,
                        float* __restrict__ out,
                        int total4 /* = N*C */)
{
    int t = blockIdx.x * blockDim.x + threadIdx.x;
    if (t >= total4) return;

    unsigned row = (unsigned)t / 7u;          // C == 7

    v4f d = __builtin_nontemporal_load(&deltas4[t]);
    v4f b = boxes4[row];

    float W1 = im_info[1] - 1.0f;
    float H1 = im_info[0] - 1.0f;

    float w  = b.z - b.x + 1.0f;
    float h  = b.w - b.y + 1.0f;
    float cx = b.x + 0.5f * w;
    float cy = b.y + 0.5f * h;

    // BBOX_NORMALIZE_STDS = [0.1, 0.1, 0.2, 0.2], means = 0
    float dx = d.x * 0.1f;
    float dy = d.y * 0.1f;
    float dw = d.z * 0.2f;
    float dh = d.w * 0.2f;

    float pcx = fmaf(dx, w, cx);
    float pcy = fmaf(dy, h, cy);
    float pw  = expf(dw) * w;
    float ph  = expf(dh) * h;

    v4f r;
    r.x = fminf(fmaxf(pcx - 0.5f * pw, 0.0f), W1);
    r.y = fminf(fmaxf(pcy - 0.5f * ph, 0.0f), H1);
    r.z = fminf(fmaxf(pcx + 0.5f * pw, 0.0f), W1);
    r.w = fminf(fmaxf(pcy + 0.5f * ph, 0.0f), H1);

    __builtin_nontemporal_store(r, (v4f*)out + t);
}

// ---------------------------------------------------------------------------
// Kernel B: per-class masked argmax partials over scores (N x 7, classes 1..6).
// Each wave async-copies a contiguous 32-row (896 B) score tile into LDS via
// gfx1250 GLOBAL_LOAD_ASYNC_TO_LDS, waits on ASYNCcnt, then each lane reads
// its stride-7 row from LDS (conflict-free: gcd(7,64)==1).
// Partials per block -> d_ws: [6 floats maxval][6 ints idx].
// ---------------------------------------------------------------------------
__global__ __launch_bounds__(256)
void score_scan_kernel(const float* __restrict__ scores,
                       int N,
                       float* __restrict__ ws)
{
    __shared__ float stage[8 * 224];      // 8 waves * 32 rows * 7 floats
    __shared__ float rv[6][256];
    __shared__ int   ri[6][256];

    const int tid   = threadIdx.x;
    const int lane  = tid & 31;
    const int wslot = tid >> 5;
    const int wavesTotal = (gridDim.x * blockDim.x) >> 5;
    const int gwave      = (blockIdx.x * blockDim.x + tid) >> 5;
    const long total = (long)N * 7;

    float bv[6];
    int   bi[6];
#pragma unroll
    for (int c = 0; c < 6; ++c) { bv[c] = -INFINITY; bi[c] = 0x7fffffff; }

    // low 32 bits of a generic pointer to LDS == wave-relative LDS byte offset
    unsigned stage_base = (unsigned)(uintptr_t)&stage[wslot * 224];

    for (long base = (long)gwave * 32; base < N; base += (long)wavesTotal * 32) {
        long tb = base * 7;
#pragma unroll
        for (int k = 0; k < 7; ++k) {
            long gi = tb + (long)(k * 32 + lane);
            if (gi >= total) gi = total - 1;          // clamp: stay in bounds
            const float* gp = scores + gi;
            unsigned la = stage_base + (unsigned)(k * 32 + lane) * 4u;
            asm volatile("global_load_async_to_lds_b32 %0, %1, off"
                         :: "v"(la), "v"(gp) : "memory");
        }
        asm volatile("s_wait_asynccnt 0x0" ::: "memory");

        long row = base + lane;
        if (row < N) {
            const float* srow = &stage[wslot * 224 + lane * 7];
            int r = (int)row;
#pragma unroll
            for (int c = 0; c < 6; ++c) {
                float s = srow[c + 1];                 // classes 1..6
                float m = (s > THRESH) ? s : -1.0f;
                if (m > bv[c] || (m == bv[c] && r < bi[c])) { bv[c] = m; bi[c] = r; }
            }
        }
    }

#pragma unroll
    for (int c = 0; c < 6; ++c) { rv[c][tid] = bv[c]; ri[c][tid] = bi[c]; }
    __syncthreads();

    for (int s = 128; s > 0; s >>= 1) {
        if (tid < s) {
#pragma unroll
            for (int c = 0; c < 6; ++c) {
                float v2 = rv[c][tid + s]; int i2 = ri[c][tid + s];
                float v1 = rv[c][tid];     int i1 = ri[c][tid];
                if (v2 > v1 || (v2 == v1 && i2 < i1)) { rv[c][tid] = v2; ri[c][tid] = i2; }
            }
        }
        __syncthreads();
    }

    if (tid == 0) {
        float* w = ws + (size_t)blockIdx.x * 12;
#pragma unroll
        for (int c = 0; c < 6; ++c) {
            w[c] = rv[c][0];
            ((int*)w)[6 + c] = ri[c][0];
        }
    }
}

// ---------------------------------------------------------------------------
// Kernel C: reduce block partials, gather winning boxes from pred (d_out),
// emit region_g (classes 1..3) and region_l (classes 4..6): 5 floats each.
// ---------------------------------------------------------------------------
__global__ __launch_bounds__(256)
void finalize_kernel(const float* __restrict__ ws, int nblocks,
                     const float* __restrict__ pred,
                     float* __restrict__ regions)
{
    __shared__ float rv[6][256];
    __shared__ int   ri[6][256];

    const int tid = threadIdx.x;
    float bv[6];
    int   bi[6];
#pragma unroll
    for (int c = 0; c < 6; ++c) { bv[c] = -INFINITY; bi[c] = 0x7fffffff; }

    for (int b = tid; b < nblocks; b += 256) {
        const float* w = ws + (size_t)b * 12;
#pragma unroll
        for (int c = 0; c < 6; ++c) {
            float v = w[c];
            int   i = ((const int*)w)[6 + c];
            if (v > bv[c] || (v == bv[c] && i < bi[c])) { bv[c] = v; bi[c] = i; }
        }
    }

#pragma unroll
    for (int c = 0; c < 6; ++c) { rv[c][tid] = bv[c]; ri[c][tid] = bi[c]; }
    __syncthreads();

    for (int s = 128; s > 0; s >>= 1) {
        if (tid < s) {
#pragma unroll
            for (int c = 0; c < 6; ++c) {
                float v2 = rv[c][tid + s]; int i2 = ri[c][tid + s];
                float v1 = rv[c][tid];     int i1 = ri[c][tid];
                if (v2 > v1 || (v2 == v1 && i2 < i1)) { rv[c][tid] = v2; ri[c][tid] = i2; }
            }
        }
        __syncthreads();
    }

    if (tid == 0) {
        // group 0: classes 1..3 (c = 0..2) -> region_g
        // group 1: classes 4..6 (c = 3..5) -> region_l
        for (int g = 0; g < 2; ++g) {
            int c0 = g * 3;
            int best = c0;
            for (int c = c0 + 1; c < c0 + 3; ++c)
                if (rv[c][0] > rv[best][0]) best = c;   // tie -> lower class idx
            int  cls = best + 1;
            long r   = ri[best][0];
            if (r == 0x7fffffff) r = 0;                 // defensive (N>0 always)
            const float* bx = pred + r * 28 + (long)cls * 4;
            float* o = regions + g * 5;
            o[0] = bx[0]; o[1] = bx[1]; o[2] = bx[2]; o[3] = bx[3];
            o[4] = rv[best][0];
        }
    }
}

// ---------------------------------------------------------------------------
extern "C" void kernel_launch(void* const* d_in, const int* in_sizes, int n_in,
                              void* d_out, int out_size, void* d_ws, size_t ws_size,
                              hipStream_t stream)
{
    const float* boxes   = (const float*)d_in[0];   // N x 4
    const float* deltas  = (const float*)d_in[1];   // N x 28
    const float* scores  = (const float*)d_in[2];   // N x 7
    const float* im_info = (const float*)d_in[3];   // 3

    const int N      = in_sizes[0] / 4;
    const int total4 = in_sizes[1] / 4;             // N * C (C == 7)
    const int C      = total4 / (N > 0 ? N : 1);

    float* out = (float*)d_out;

    // Phase A: decode + clip (bandwidth-bound streaming)
    {
        int blk = 256;
        int grd = (total4 + blk - 1) / blk;
        decode_clip_kernel<<<grd, blk, 0, stream>>>(
            (const v4f*)deltas, (const v4f*)boxes, im_info, out, total4);
    }

    // Phase B: per-class masked argmax partials
    size_t maxb = ws_size / (12 * sizeof(float));
    int nb = (int)(maxb < 1024 ? maxb : 1024);
    if (nb < 1) nb = 1;
    score_scan_kernel<<<nb, 256, 0, stream>>>(scores, N, (float*)d_ws);

    // Phase C: finalize regions (reads pred from d_out written in phase A)
    float* regions = out + (size_t)N * C * 4;
    finalize_kernel<<<1, 256, 0, stream>>>((const float*)d_ws, nb, out, regions);
}